// PostProcess_74199855005721
// MI455X (gfx1250) — compile-verified
//
#include <hip/hip_runtime.h>
#include <hip/hip_bf16.h>

typedef unsigned long long ull;
typedef float v2f __attribute__((ext_vector_type(2)));
typedef float v8f __attribute__((ext_vector_type(8)));

#define IMGB   32
#define NANCH  22743
#define NPAD   32768
#define KTOP   2048
#define NCLS   80

#if defined(__AMDGCN__) && __has_builtin(__builtin_amdgcn_global_load_async_to_lds_b32)
#define HAVE_ASYNC_LDS 1
typedef __attribute__((address_space(1))) int async_gint;  // global src, b32 units
typedef __attribute__((address_space(3))) int async_lint;  // LDS dst, b32 units
#else
#define HAVE_ASYNC_LDS 0
#endif

__device__ __forceinline__ void wait_asynccnt0() {
#if defined(__AMDGCN__)
#if __has_builtin(__builtin_amdgcn_s_wait_asynccnt)
  __builtin_amdgcn_s_wait_asynccnt(0);
#else
  asm volatile("s_wait_asynccnt 0" ::: "memory");
#endif
#endif
}

// ---------------------------------------------------------------------------
// Phase 1: stream features (247MB — the only HBM-roofline phase, ~11us floor at
// 23.3 TB/s), compute score = max(cls)*obj, argmax label, build sortable key
//   u64 key = (~orderable(masked_score) << 32) | anchor_idx.
// Data movement: GLOBAL_LOAD_ASYNC_TO_LDS_B32 (ASYNCcnt-tracked, no VGPR
// staging) + one s_wait_asynccnt 0, plus global_prefetch for the next tile.
// ---------------------------------------------------------------------------
__global__ void score_kernel(const float* __restrict__ feat,
                             ull* __restrict__ keys,
                             int* __restrict__ labels_all,
                             const float* __restrict__ score_thresh) {
  __shared__ float tile[128 * 85];
  const int b  = blockIdx.y;
  const int n0 = blockIdx.x * 128;
  const int t  = threadIdx.x;
  const int n  = n0 + t;

  const int cnt = (n0 < NANCH) ? min(128, NANCH - n0) : 0;
  if (cnt > 0) {
    const float* src = feat + ((size_t)b * NANCH + n0) * 85;
    // prefetch the next block's stream (speculative, L2)
    __builtin_prefetch(src + 128 * 85 + t * 32, 0, 1);
    const int tot = cnt * 85;
#if HAVE_ASYNC_LDS
    // deep async pipeline: up to 85 outstanding copies/thread; HW stalls issue
    // if ASYNCcnt (6-bit) would overflow, so no manual throttling needed.
    for (int i = t; i < tot; i += 128)
      __builtin_amdgcn_global_load_async_to_lds_b32(
          (async_gint*)(src + i), (async_lint*)(&tile[i]), 0, 0);
    wait_asynccnt0();
#else
    for (int i = t; i < tot; i += 128) tile[i] = src[i];
#endif
  }
  __syncthreads();

  const float thr = score_thresh[0];
  ull key;
  if (t < cnt) {
    const float* row = &tile[t * 85];
    const float obj = row[4];
    float best = row[5];
    int bi = 0;
#pragma unroll
    for (int j = 1; j < NCLS; j++) {
      float c = row[5 + j];
      if (c > best) { best = c; bi = j; }   // first-max tie-break == jnp.argmax
    }
    const float s = best * obj;
    const float m = (s >= thr) ? s : -1.0f;
    unsigned bits = __float_as_uint(m);
    unsigned u    = (bits & 0x80000000u) ? ~bits : (bits | 0x80000000u); // total order
    key = ((ull)(~u) << 32) | (unsigned)n;   // ascending sort => score desc, idx asc
    labels_all[(size_t)b * NANCH + n] = bi;
  } else {
    key = 0xFFFFFFFF00000000ULL | (unsigned)n; // padding sorts last
  }
  if (n < NPAD) keys[(size_t)b * NPAD + n] = key;
}

// ---------------------------------------------------------------------------
// Phase 2: per-image bitonic sort of 32768 u64 keys (L2-resident, one block per
// image). Ascending key == exact lax.top_k ordering (stable desc, idx-asc ties).
// ---------------------------------------------------------------------------
__global__ void sort_kernel(ull* __restrict__ keys) {
  ull* kb = keys + (size_t)blockIdx.x * NPAD;
  const int n = NPAD;
  const int tid = threadIdx.x;
  for (int k = 2; k <= n; k <<= 1) {
    for (int j = k >> 1; j > 0; j >>= 1) {
      for (int i = tid; i < n; i += 1024) {
        const int ij = i ^ j;
        if (ij > i) {
          ull a = kb[i], c = kb[ij];
          const bool up = ((i & k) == 0);
          if ((a > c) == up) { kb[i] = c; kb[ij] = a; }
        }
      }
      __syncthreads();
    }
  }
}

__device__ __forceinline__ float decode_score(unsigned keyhi) {
  const unsigned u = ~keyhi;
  return (u >> 31) ? __uint_as_float(u ^ 0x80000000u) : __uint_as_float(~u);
}

// ---------------------------------------------------------------------------
// Phase 3: gather top-K, decode+clip boxes, areas, labels, valid-prefix count v.
// ---------------------------------------------------------------------------
__global__ void prep_kernel(const ull* __restrict__ keys,
                            const float* __restrict__ feat,
                            const float* __restrict__ anchors,
                            const int* __restrict__ labels_all,
                            const float* __restrict__ image_sizes,
                            const float* __restrict__ score_thresh,
                            float* __restrict__ top_score,
                            int* __restrict__ lab_k,
                            float4* __restrict__ boxes4,
                            float* __restrict__ area,
                            int* __restrict__ vcount) {
  const int t = blockIdx.x * blockDim.x + threadIdx.x;
  if (t >= IMGB * KTOP) return;
  const int b = t / KTOP, k = t % KTOP;
  const ull* kb = keys + (size_t)b * NPAD;
  const ull key = kb[k];
  const unsigned idx = (unsigned)(key & 0xFFFFFFFFu);
  const float s = decode_score((unsigned)(key >> 32));
  const float thr = score_thresh[0];
  const bool valid = (s >= thr);

  // valid entries form a prefix (sorted); exactly one thread writes v.
  if (k == 0 && !valid) vcount[b] = 0;
  if (valid) {
    if (k == KTOP - 1) vcount[b] = KTOP;
    else {
      const float s2 = decode_score((unsigned)(kb[k + 1] >> 32));
      if (!(s2 >= thr)) vcount[b] = k + 1;
    }
  }

  const float* fr = feat + ((size_t)b * NANCH + idx) * 85;
  const float* an = anchors + (size_t)idx * 5;
  const float cx = (fr[0] + an[0]) * an[2];
  const float cy = (fr[1] + an[1]) * an[2];
  const float w  = an[3] * expf(fr[2]);
  const float h  = an[4] * expf(fr[3]);
  float l = cx - 0.5f * w, tp = cy - 0.5f * h;
  float r = l + w, bm = tp + h;
  const float H = image_sizes[b * 2 + 0], W = image_sizes[b * 2 + 1];
  l  = fminf(fmaxf(l, 0.f), W - 1.f);  r  = fminf(fmaxf(r, 0.f), W - 1.f);
  tp = fminf(fmaxf(tp, 0.f), H - 1.f); bm = fminf(fmaxf(bm, 0.f), H - 1.f);

  top_score[t] = s;
  lab_k[t] = labels_all[(size_t)b * NANCH + idx];
  boxes4[t] = make_float4(l, tp, r, bm);
  area[t]   = (r - l) * (bm - tp);
}

// ---------------------------------------------------------------------------
// Phase 4: tiled suppression matrix S[i][j] = (label_i==label_j) & (IoU>thr).
// One wave per 16-row strip; only tiles with i<v AND j<v are computed.
// area_i + area_j per 16x16 tile via V_WMMA_F32_16X16X4_F32 (rank-2 product);
// wave32 ballot packs the 16-bit suppression rows (low16 = M=r, high16 = M=r+8,
// matching the C/D VGPR layout).
// ---------------------------------------------------------------------------
__global__ void iou_tiles_kernel(const float4* __restrict__ boxes4,
                                 const float* __restrict__ area,
                                 const int* __restrict__ lab_k,
                                 const int* __restrict__ vcount,
                                 const float* __restrict__ nms_thresh,
                                 unsigned short* __restrict__ S) {
  const int b    = blockIdx.y;
  const int wv   = threadIdx.x >> 5;
  const int lane = threadIdx.x & 31;
  const int I    = blockIdx.x * 8 + wv;          // tile-row 0..127 (wave-uniform)
  const int v    = vcount[b];
  if (I * 16 >= v) return;                        // uniform exit: EXEC stays all-1s
  const int Jmax = (v + 15) >> 4;
  const float nms = nms_thresh[0];

  const float4* bb = boxes4 + (size_t)b * KTOP;
  const float*  ar = area   + (size_t)b * KTOP;
  const int*    lb = lab_k  + (size_t)b * KTOP;
  const int i0 = I * 16;

  // A (16x4 f32): lanes 0-15 carry K0,K1 = (area_i[M=lane], 1); lanes 16-31 K2,K3 = 0
  v2f afrag;
  afrag.x = (lane < 16) ? ar[i0 + lane] : 0.f;
  afrag.y = (lane < 16) ? 1.f : 0.f;

  const int mBase = (lane < 16) ? 0 : 8;          // D: vgpr r -> M=r (lo lanes) / r+8 (hi)
  float4 bi[8]; int li[8];
#pragma unroll
  for (int r = 0; r < 8; r++) { bi[r] = bb[i0 + mBase + r]; li[r] = lb[i0 + mBase + r]; }

  for (int J = 0; J < Jmax; J++) {
    const int j0 = J * 16;
    const int nj = j0 + (lane & 15);
    // B (4x16 f32): lanes 0-15 carry K0,K1 = (1, area_j[N=lane]); lanes 16-31 = 0
    v2f bfrag;
    bfrag.x = (lane < 16) ? 1.f : 0.f;
    bfrag.y = (lane < 16) ? ar[j0 + lane] : 0.f;
    v8f cz = {0.f, 0.f, 0.f, 0.f, 0.f, 0.f, 0.f, 0.f};
    // d[r] = area_i[M] + area_j[N]  (rank-2 outer product)
    v8f d = __builtin_amdgcn_wmma_f32_16x16x4_f32(false, afrag, false, bfrag,
                                                  (short)0, cz, false, false);
    const float4 bj = bb[nj];
    const int    lj = lb[nj];
    unsigned rows[8];
#pragma unroll
    for (int r = 0; r < 8; r++) {
      const float x1 = fmaxf(bi[r].x, bj.x), y1 = fmaxf(bi[r].y, bj.y);
      const float x2 = fminf(bi[r].z, bj.z), y2 = fminf(bi[r].w, bj.w);
      const float iw = fmaxf(x2 - x1, 0.f),  ih = fmaxf(y2 - y1, 0.f);
      const float inter = iw * ih;
      const float iou = inter / (d[r] - inter + 1e-9f);
      const bool pred = (iou > nms) && (li[r] == lj);  // label-offset trick == equality
      rows[r] = __builtin_amdgcn_ballot_w32(pred);
    }
    if (lane == 0) {
#pragma unroll
      for (int r = 0; r < 8; r++) {
        S[((size_t)b * KTOP + i0 + r)     * 128 + J] = (unsigned short)(rows[r] & 0xFFFFu);
        S[((size_t)b * KTOP + i0 + r + 8) * 128 + J] = (unsigned short)(rows[r] >> 16);
      }
    }
  }
}

// ---------------------------------------------------------------------------
// Phase 5: sequential greedy resolution — exact reference recurrence.
// One wave per image; active mask (64 u32) in LDS; active &= ~S[i] (self bit
// excluded). Only words < ceil(v/32) touched => unwritten S garbage is never
// applied to a live column.
// ---------------------------------------------------------------------------
__global__ void resolve_kernel(const float* __restrict__ top_score,
                               const float* __restrict__ score_thresh,
                               const int* __restrict__ vcount,
                               const unsigned short* __restrict__ S,
                               unsigned* __restrict__ active_out) {
  __shared__ unsigned act[64];
  const int b = blockIdx.x;
  const int lane = threadIdx.x;                   // 32 threads
  const float thr = score_thresh[0];
  const float* ts = top_score + (size_t)b * KTOP;
  for (int w = lane; w < 64; w += 32) {
    unsigned m = 0;
    for (int t2 = 0; t2 < 32; t2++)
      if (ts[w * 32 + t2] >= thr) m |= (1u << t2);
    act[w] = m;
  }
  __syncthreads();
  const int v = vcount[b];
  const int wmax = (v + 31) >> 5;
  const unsigned* Sb = (const unsigned*)(S + (size_t)b * KTOP * 128);
  for (int i = 0; i < v; i++) {
    __syncthreads();
    const bool on = (act[i >> 5] >> (i & 31)) & 1; // uniform across wave
    if (on) {
      const unsigned* row = Sb + (size_t)i * 64;
#pragma unroll
      for (int q = 0; q < 2; q++) {
        const int w = lane + q * 32;
        if (w < wmax) {
          unsigned m = row[w];
          if (w == (i >> 5)) m &= ~(1u << (i & 31)); // suppress.at[i] = False
          act[w] &= ~m;
        }
      }
    }
  }
  __syncthreads();
  for (int w = lane; w < 64; w += 32) active_out[(size_t)b * 64 + w] = act[w];
}

// ---------------------------------------------------------------------------
// Phase 6: emit (boxes*ratio, scores, labels, keep) concatenated flat.
// ---------------------------------------------------------------------------
__global__ void output_kernel(const float4* __restrict__ boxes4,
                              const float* __restrict__ top_score,
                              const int* __restrict__ lab_k,
                              const unsigned* __restrict__ active_out,
                              const float* __restrict__ image_sizes,
                              const float* __restrict__ image_sizes_ori,
                              float* __restrict__ out) {
  const int t = blockIdx.x * blockDim.x + threadIdx.x;
  if (t >= IMGB * KTOP) return;
  const int b = t / KTOP, k = t % KTOP;
  const float ratio = image_sizes_ori[b * 2] / image_sizes[b * 2];
  const float4 bx = boxes4[t];
  float* ob = out + (size_t)t * 4;
  ob[0] = bx.x * ratio; ob[1] = bx.y * ratio;
  ob[2] = bx.z * ratio; ob[3] = bx.w * ratio;
  const size_t base = (size_t)IMGB * KTOP;
  out[base * 4 + t] = top_score[t];
  out[base * 5 + t] = (float)lab_k[t];
  const unsigned aw = active_out[(size_t)b * 64 + (k >> 5)];
  out[base * 6 + t] = ((aw >> (k & 31)) & 1) ? 1.0f : 0.0f;
}

extern "C" void kernel_launch(void* const* d_in, const int* in_sizes, int n_in,
                              void* d_out, int out_size, void* d_ws, size_t ws_size,
                              hipStream_t stream) {
  const float* feat         = (const float*)d_in[0];
  const float* anchors      = (const float*)d_in[1];
  const float* image_sizes  = (const float*)d_in[2];
  const float* image_sz_ori = (const float*)d_in[3];
  const float* score_thresh = (const float*)d_in[4];
  const float* nms_thresh   = (const float*)d_in[5];
  float* out = (float*)d_out;

  // workspace carve-up (256B aligned), total ~29 MB
  char* ws = (char*)d_ws;
  size_t off = 0;
  auto carve = [&](size_t bytes) {
    char* p = ws + off;
    off = (off + bytes + 255) & ~(size_t)255;
    return p;
  };
  ull*      keys       = (ull*)      carve((size_t)IMGB * NPAD * 8);
  int*      labels_all = (int*)      carve((size_t)IMGB * NANCH * 4);
  float*    top_score  = (float*)    carve((size_t)IMGB * KTOP * 4);
  int*      lab_k      = (int*)      carve((size_t)IMGB * KTOP * 4);
  float4*   boxes4     = (float4*)   carve((size_t)IMGB * KTOP * 16);
  float*    area       = (float*)    carve((size_t)IMGB * KTOP * 4);
  int*      vcount     = (int*)      carve((size_t)IMGB * 4);
  unsigned* active_out = (unsigned*) carve((size_t)IMGB * 64 * 4);
  unsigned short* S    = (unsigned short*)carve((size_t)IMGB * KTOP * 128 * 2);
  (void)ws_size; (void)in_sizes; (void)n_in; (void)out_size;

  score_kernel<<<dim3(NPAD / 128, IMGB), 128, 0, stream>>>(feat, keys, labels_all, score_thresh);
  sort_kernel<<<IMGB, 1024, 0, stream>>>(keys);
  prep_kernel<<<(IMGB * KTOP + 255) / 256, 256, 0, stream>>>(
      keys, feat, anchors, labels_all, image_sizes, score_thresh,
      top_score, lab_k, boxes4, area, vcount);
  iou_tiles_kernel<<<dim3(16, IMGB), 256, 0, stream>>>(boxes4, area, lab_k, vcount, nms_thresh, S);
  resolve_kernel<<<IMGB, 32, 0, stream>>>(top_score, score_thresh, vcount, S, active_out);
  output_kernel<<<(IMGB * KTOP + 255) / 256, 256, 0, stream>>>(
      boxes4, top_score, lab_k, active_out, image_sizes, image_sz_ori, out);
}